// MPred_72773925864254
// MI455X (gfx1250) — compile-verified
//
#include <hip/hip_runtime.h>

// YOLO decode: out[..., 0:2]=sigmoid, [...,2:4]=exp/img, [...,4]=sigmoid, rest=copy.
// Pure streaming elementwise op: 231.7 MB traffic -> ~10 us floor at 23.3 TB/s.
// Strategy: 4x 128-bit coalesced non-temporal loads/stores per thread
// (block-strided so each b128 is contiguous across the wave), single
// v_exp_f32 per element, branchless channel select.

typedef float v4f __attribute__((ext_vector_type(4)));

#define CH     85
#define CHUNKS 4
#define BLOCK  256

__device__ __forceinline__ v4f decode_chunk(v4f v, int i, float inv_img)
{
    // Channel of first element of this float4. Elements j=0..3 are channels
    // c0+j with a single wrap at CH (c0 <= 84 -> c0+3 <= 87).
    const int c0 = (i << 2) % CH;

    float r[4] = { v.x, v.y, v.z, v.w };

#pragma unroll
    for (int j = 0; j < 4; ++j) {
        int ch = c0 + j;
        ch = (ch >= CH) ? ch - CH : ch;

        const float xx = r[j];

        // One transcendental per element:
        //   t = exp(x);  sigmoid = t/(1+t);  wh = t/img_size
        const float t   = __expf(xx);
        const float sig = t * __builtin_amdgcn_rcpf(1.0f + t);
        const float wh  = t * inv_img;

        const bool is_sig = (ch < 2) || (ch == 4);   // x, y, obj
        const bool is_wh  = (ch == 2) || (ch == 3);  // w, h

        r[j] = is_sig ? sig : (is_wh ? wh : xx);
    }

    v4f o;
    o.x = r[0]; o.y = r[1]; o.z = r[2]; o.w = r[3];
    return o;
}

__global__ void __launch_bounds__(BLOCK)
yolo_decode_kernel(const float* __restrict__ x,
                   const int*   __restrict__ img_size,
                   float*       __restrict__ out,
                   int n4)
{
    const int tid  = threadIdx.x;
    const int base = blockIdx.x * (BLOCK * CHUNKS);

    // Uniform: 1/img_size via native rcp (1 ulp; exp error dominates anyway).
    const float inv_img = __builtin_amdgcn_rcpf((float)img_size[0]);

    const v4f* __restrict__ xv = (const v4f*)x;
    v4f*       __restrict__ ov = (v4f*)out;

    if (base + BLOCK * CHUNKS <= n4) {
        // Fast path: no per-chunk bounds checks -> loads cluster, latencies overlap.
        v4f v[CHUNKS];
#pragma unroll
        for (int k = 0; k < CHUNKS; ++k)
            v[k] = __builtin_nontemporal_load(&xv[base + k * BLOCK + tid]);
#pragma unroll
        for (int k = 0; k < CHUNKS; ++k) {
            const int i = base + k * BLOCK + tid;
            __builtin_nontemporal_store(decode_chunk(v[k], i, inv_img), &ov[i]);
        }
    } else {
        // Tail block: guard each chunk.
#pragma unroll
        for (int k = 0; k < CHUNKS; ++k) {
            const int i = base + k * BLOCK + tid;
            if (i < n4) {
                v4f v = __builtin_nontemporal_load(&xv[i]);
                __builtin_nontemporal_store(decode_chunk(v, i, inv_img), &ov[i]);
            }
        }
    }
}

extern "C" void kernel_launch(void* const* d_in, const int* in_sizes, int n_in,
                              void* d_out, int out_size, void* d_ws, size_t ws_size,
                              hipStream_t stream)
{
    const float* x        = (const float*)d_in[0];   // (32, 10647, 85) fp32
    const int*   img_size = (const int*)d_in[1];     // scalar int (1 element)
    float*       out      = (float*)d_out;

    const int n  = in_sizes[0];        // 28,959,840 (divisible by 4)
    const int n4 = n >> 2;             // float4 chunks

    const int per_block = BLOCK * CHUNKS;
    const int grid      = (n4 + per_block - 1) / per_block;

    yolo_decode_kernel<<<grid, BLOCK, 0, stream>>>(x, img_size, out, n4);
}